// BA_TripletLoss_37812892074051
// MI455X (gfx1250) — compile-verified
//
#include <hip/hip_runtime.h>
#include <hip/hip_bf16.h>

typedef __attribute__((ext_vector_type(2))) float v2f;
typedef __attribute__((ext_vector_type(8))) float v8f;

#define MARGIN 200.0f
#define EPS_CLAMP 1e-12f
#define EPS_VALID 1e-9f
#define MAXN 1024

// -------------------------------------------------------------------------
// Kernel 1: Gram matrix G = X * X^T via V_WMMA_F32_16X16X4_F32.
// One wave32 computes one 16x16 tile, accumulating over K in steps of 4.
// f32 A-matrix 16x4 layout (ISA 7.12.2): lanes 0-15 hold M=0..15 with
// v0=K0,v1=K1; lanes 16-31 hold v0=K2,v1=K3. B (4x16) mirrors with columns.
// C/D layout: VGPR r -> (M=r, N=lane) for lanes 0-15, (M=8+r, N=lane-16).
// -------------------------------------------------------------------------
__global__ __launch_bounds__(256) void gram_wmma_kernel(
    const float* __restrict__ X, float* __restrict__ G, int n, int d) {
  const int wave = (blockIdx.x * blockDim.x + threadIdx.x) >> 5;  // wave-uniform
  const int lane = threadIdx.x & 31;
  const int tilesPerRow = n >> 4;
  const int totalTiles = tilesPerRow * tilesPerRow;
  if (wave >= totalTiles) return;  // wave-uniform branch: EXEC stays all-1s

  const int tm = (wave / tilesPerRow) << 4;
  const int tn = (wave % tilesPerRow) << 4;

  const int hl = lane >> 4;     // 0: K={0,1}, 1: K={2,3}
  const int l15 = lane & 15;
  const float* __restrict__ arow = X + (size_t)(tm + l15) * d + 2 * hl;
  const float* __restrict__ brow = X + (size_t)(tn + l15) * d + 2 * hl;

  v8f acc = {};
#pragma unroll 4
  for (int k = 0; k < d; k += 4) {
    v2f a = *(const v2f*)(arow + k);   // 8B aligned: k even, row base 8B aligned
    v2f b = *(const v2f*)(brow + k);
    acc = __builtin_amdgcn_wmma_f32_16x16x4_f32(
        /*neg_a=*/false, a, /*neg_b=*/false, b,
        /*c_mod=*/(short)0, acc, /*reuse_a=*/false, /*reuse_b=*/false);
  }

#pragma unroll
  for (int r = 0; r < 8; ++r) {
    G[(size_t)(tm + r + 8 * hl) * n + (tn + l15)] = acc[r];
  }
}

// -------------------------------------------------------------------------
// Kernel 2: pull diagonal (sq norms) out before dist overwrites G in place.
// -------------------------------------------------------------------------
__global__ void diag_kernel(const float* __restrict__ G,
                            float* __restrict__ diag, int n) {
  int i = blockIdx.x * blockDim.x + threadIdx.x;
  if (i < n) diag[i] = G[(size_t)i * n + i];
}

// -------------------------------------------------------------------------
// Kernel 3: dist[i,j] = max(diag[i] + diag[j] - 2 G[i,j], 1e-12), in place.
// -------------------------------------------------------------------------
__global__ void dist_kernel(float* __restrict__ G,
                            const float* __restrict__ diag, int n) {
  int idx = blockIdx.x * blockDim.x + threadIdx.x;
  if (idx >= n * n) return;
  int i = idx / n;
  int j = idx - i * n;
  float dv = diag[i] + diag[j] - 2.0f * G[idx];
  G[idx] = fmaxf(dv, EPS_CLAMP);
}

// -------------------------------------------------------------------------
// Kernel 4: per-anchor batch-all hinge. One block per anchor i; dist row and
// negative mask live in LDS; inner loop is branch-free mask*max FMA chain.
// Writes a per-anchor partial (no float atomics -> deterministic).
// -------------------------------------------------------------------------
__global__ __launch_bounds__(256) void triplet_kernel(
    const float* __restrict__ dist, const int* __restrict__ targets,
    float* __restrict__ partial, int n) {
  __shared__ float drow[MAXN];
  __shared__ float negm[MAXN];
  __shared__ int tgt[MAXN];
  __shared__ float red[256];

  const int i = blockIdx.x;
  const int tid = threadIdx.x;
  const int ti = targets[i];

  for (int j = tid; j < n; j += blockDim.x) {
    float dv = dist[(size_t)i * n + j];
    int tj = targets[j];
    drow[j] = dv;
    tgt[j] = tj;
    negm[j] = (tj != ti) ? 1.0f : 0.0f;
  }
  __syncthreads();

  float sum = 0.0f;
  for (int p = tid; p < n; p += blockDim.x) {
    if (tgt[p] == ti && drow[p] > EPS_VALID) {
      const float a = drow[p] + MARGIN;
#pragma unroll 4
      for (int k = 0; k < n; ++k) {
        sum += negm[k] * fmaxf(a - drow[k], 0.0f);
      }
    }
  }

  red[tid] = sum;
  __syncthreads();
#pragma unroll
  for (int s = 128; s > 0; s >>= 1) {
    if (tid < s) red[tid] += red[tid + s];
    __syncthreads();
  }
  if (tid == 0) partial[i] = red[0];
}

// -------------------------------------------------------------------------
// Kernel 5: finalize. Fixed-order reductions (deterministic):
//  - loss_sum = sum(partial)
//  - count    = #(valid anchor-positive pairs)
//  - last_i   = largest anchor with any valid positive
//  - neg_pairs = #negatives of that anchor; out = loss / (count * neg_pairs)
// -------------------------------------------------------------------------
__global__ __launch_bounds__(256) void finalize_kernel(
    const float* __restrict__ dist, const int* __restrict__ targets,
    const float* __restrict__ partial, float* __restrict__ out, int n) {
  __shared__ float fred[256];
  __shared__ int cred[256];
  __shared__ int lred[256];
  const int tid = threadIdx.x;

  float ls = 0.0f;
  for (int i = tid; i < n; i += 256) ls += partial[i];

  int cnt = 0;
  for (int idx = tid; idx < n * n; idx += 256) {
    int i = idx / n;
    int j = idx - i * n;
    if (targets[i] == targets[j] && dist[idx] > EPS_VALID) cnt++;
  }

  int lastLocal = -1;
  for (int i = tid; i < n; i += 256) {
    bool hv = false;
    int ti = targets[i];
    for (int j = 0; j < n; ++j) {
      if (targets[j] == ti && dist[(size_t)i * n + j] > EPS_VALID) { hv = true; break; }
    }
    if (hv) lastLocal = i;  // i strictly increasing per thread
  }

  fred[tid] = ls;
  cred[tid] = cnt;
  lred[tid] = lastLocal;
  __syncthreads();
#pragma unroll
  for (int s = 128; s > 0; s >>= 1) {
    if (tid < s) {
      fred[tid] += fred[tid + s];
      cred[tid] += cred[tid + s];
      lred[tid] = max(lred[tid], lred[tid + s]);
    }
    __syncthreads();
  }

  if (tid == 0) {
    int last_i = lred[0] < 0 ? (n - 1) : lred[0];
    int tlast = targets[last_i];
    int neg = 0;
    for (int j = 0; j < n; ++j)
      if (targets[j] != tlast) neg++;
    float denom = (float)cred[0] * (float)neg;
    out[0] = fred[0] / denom;
  }
}

// -------------------------------------------------------------------------
extern "C" void kernel_launch(void* const* d_in, const int* in_sizes, int n_in,
                              void* d_out, int out_size, void* d_ws, size_t ws_size,
                              hipStream_t stream) {
  const float* X = (const float*)d_in[0];       // [n, d] fp32
  const int* targets = (const int*)d_in[1];     // [n] int
  float* out = (float*)d_out;

  const int n = in_sizes[1];          // 512
  const int d = in_sizes[0] / n;      // 2048

  // Workspace layout: G/dist [n*n] | diag [n] | partial [n]
  float* G = (float*)d_ws;
  float* diag = G + (size_t)n * n;
  float* partial = diag + n;

  // 1) Gram matrix via WMMA: one wave per 16x16 tile
  {
    int tiles = (n / 16) * (n / 16);
    int threads = tiles * 32;
    int blocks = (threads + 255) / 256;
    gram_wmma_kernel<<<blocks, 256, 0, stream>>>(X, G, n, d);
  }
  // 2) diagonal
  diag_kernel<<<(n + 255) / 256, 256, 0, stream>>>(G, diag, n);
  // 3) pairwise distances (in place over G)
  dist_kernel<<<((size_t)n * n + 255) / 256, 256, 0, stream>>>(G, diag, n);
  // 4) per-anchor hinge partials
  triplet_kernel<<<n, 256, 0, stream>>>(G, targets, partial, n);
  // 5) finalize scalar
  finalize_kernel<<<1, 256, 0, stream>>>(G, targets, partial, out, n);
}